// MatrixHyperlayer_46634754900206
// MI455X (gfx1250) — compile-verified
//
#include <hip/hip_runtime.h>

#define IN_NUM   50000
#define OUT_NUM  50000
#define KTOT     131072
#define DFEAT    64

typedef float v2f __attribute__((ext_vector_type(2)));
typedef float v8f __attribute__((ext_vector_type(8)));

__device__ __forceinline__ float sigmoidf_(float v) {
  return 1.0f / (1.0f + __expf(-v));
}
__device__ __forceinline__ float softplusf_(float v) {
  // jax.nn.softplus = log1p(exp(v)); stable for large v
  return (v > 20.0f) ? v : log1pf(__expf(v));
}

__global__ void MH_zero_kernel(float4* __restrict__ p, int n4) {
  int i = blockIdx.x * blockDim.x + threadIdx.x;
  if (i < n4) p[i] = make_float4(0.f, 0.f, 0.f, 0.f);
}

// One wave (32 lanes) handles 8 consecutive k's:
//   - lanes (4 per k) compute the 8 Gaussian props (4 neighbor + 4 sampled)
//   - neighbor part: 16x16 block-diag weight matrix (2x2 blocks) x 16 x-rows
//     via V_WMMA_F32_16X16X4_F32, then 16 row-atomic updates (2 per k)
//   - sampled part: broadcast loop, coalesced f32 atomics (4 rows per k)
__global__ __launch_bounds__(256) void MH_hyper_kernel(
    const float* __restrict__ x, const float* __restrict__ params,
    const int* __restrict__ sints, float* __restrict__ out)
{
  __shared__ float ldsA[8][16][16];   // per-wave 16x16 A matrix
  __shared__ int   ldsRow[8][16];     // per-wave output-row per slot
  __shared__ int   ldsCol[8][16];     // per-wave x-row per K slot

  const int lane   = threadIdx.x & 31;
  const int wv     = threadIdx.x >> 5;
  const int waveId = blockIdx.x * 8 + wv;
  const int klocal = lane >> 2;       // 0..7 : k within wave batch
  const int j      = lane & 3;        // 0..3 : neighbor/sample index
  const int k      = waveId * 8 + klocal;

  // zero this wave's A tile (256 floats, 8 per lane)
  float* Aflat = &ldsA[wv][0][0];
#pragma unroll
  for (int t = 0; t < 8; ++t) Aflat[lane * 8 + t] = 0.0f;

  // ---- per-k Gaussian weight math (4 lanes per k, redundant scalars) ----
  const float4 p = reinterpret_cast<const float4*>(params)[k];
  const float m0 = sigmoidf_(p.x) * (float)(OUT_NUM - 1);
  const float m1 = sigmoidf_(p.y) * (float)(IN_NUM - 1);
  const float sp = softplusf_(p.z + 2.0f) + 1e-6f;           // SIGMA_BOOST
  const float inv0 = rsqrtf(1e-6f + sp * ((float)OUT_NUM * 0.2f));
  const float inv1 = rsqrtf(1e-6f + sp * ((float)IN_NUM  * 0.2f));

  const float fr = floorf(m0), cr = ceilf(m0);
  const float fc = floorf(m1), cc = ceilf(m1);
  // FLOOR_MASK rows: j=0:(fr,fc) j=1:(fr,cc) j=2:(cr,fc) j=3:(cr,cc)
  const float nr = (j < 2) ? fr : cr;
  const float nc = (j & 1) ? cc : fc;

  float d0 = (nr - m0) * inv0;
  float d1 = (nc - m1) * inv1;
  const float pn = __expf(-0.5f * (d0 * d0 + d1 * d1));

  const int2 si = reinterpret_cast<const int2*>(sints)[k * 4 + j];
  d0 = ((float)si.x - m0) * inv0;
  d1 = ((float)si.y - m1) * inv1;
  const float ps = __expf(-0.5f * (d0 * d0 + d1 * d1));

  // normalizer over the 8 props of this k (groups of 4 lanes)
  float tot = pn + ps;
  tot += __shfl_xor(tot, 1, 32);
  tot += __shfl_xor(tot, 2, 32);
  const float scale = p.w / (tot + 8.0f * 1e-6f);   // sum(props + eps)
  const float wn    = pn * scale;
  const float wsamp = ps * scale;

  // ---- build block-diagonal A and slot->row/col tables in LDS ----
  ldsA[wv][2 * klocal + (j >> 1)][2 * klocal + (j & 1)] = wn;
  if (j == 0) {
    ldsRow[wv][2 * klocal]     = (int)fr;
    ldsRow[wv][2 * klocal + 1] = (int)cr;
    ldsCol[wv][2 * klocal]     = (int)fc;
    ldsCol[wv][2 * klocal + 1] = (int)cc;
  }
  // same-wave DS ordering; make results visible before readback
  asm volatile("s_wait_dscnt 0" ::: "memory");

  // ---- neighbor combine: D(16x64) = A(16x16) @ B(16x64) via WMMA f32 ----
  const int nsel  = lane & 15;   // N index (feature within 16-wide tile)
  const int khalf = lane >> 4;   // lane-half: K offset of 2 per ISA layout

  v8f acc0 = {}, acc1 = {}, acc2 = {}, acc3 = {};
#pragma unroll
  for (int i = 0; i < 4; ++i) {       // K chunks of 4
    const int r0 = 4 * i + 2 * khalf;
    const int c0 = ldsCol[wv][r0];
    const int c1 = ldsCol[wv][r0 + 1];
    v2f a;                             // A: lane=M, VGPR0/1 = K r0,r0+1
    a.x = ldsA[wv][nsel][r0];
    a.y = ldsA[wv][nsel][r0 + 1];
    const float* xr0 = x + (size_t)c0 * DFEAT + nsel;
    const float* xr1 = x + (size_t)c1 * DFEAT + nsel;
    v2f b;                             // B: lane=N, VGPR0/1 = K r0,r0+1
    b.x = xr0[0];  b.y = xr1[0];
    acc0 = __builtin_amdgcn_wmma_f32_16x16x4_f32(false, a, false, b, (short)0, acc0, false, false);
    b.x = xr0[16]; b.y = xr1[16];
    acc1 = __builtin_amdgcn_wmma_f32_16x16x4_f32(false, a, false, b, (short)0, acc1, false, false);
    b.x = xr0[32]; b.y = xr1[32];
    acc2 = __builtin_amdgcn_wmma_f32_16x16x4_f32(false, a, false, b, (short)0, acc2, false, false);
    b.x = xr0[48]; b.y = xr1[48];
    acc3 = __builtin_amdgcn_wmma_f32_16x16x4_f32(false, a, false, b, (short)0, acc3, false, false);
  }

  // scatter the 16 combined rows (2 per k) with coalesced f32 atomics
  int grow[8];
#pragma unroll
  for (int v = 0; v < 8; ++v) grow[v] = ldsRow[wv][v + 8 * khalf];
#pragma unroll
  for (int v = 0; v < 8; ++v) {
    float* po = out + (size_t)grow[v] * DFEAT + nsel;
    atomicAdd(po +  0, acc0[v]);
    atomicAdd(po + 16, acc1[v]);
    atomicAdd(po + 32, acc2[v]);
    atomicAdd(po + 48, acc3[v]);
  }

  // ---- sampled sites: broadcast one (row,col,w) per iteration ----
  const float2* x2 = reinterpret_cast<const float2*>(x);
#pragma unroll 1
  for (int src = 0; src < 32; ++src) {
    const int   srow = __shfl(si.x, src, 32);
    const int   scol = __shfl(si.y, src, 32);
    const float wq   = __shfl(wsamp, src, 32);
    const float2 xv  = x2[(size_t)scol * (DFEAT / 2) + lane];
    float* po = out + (size_t)srow * DFEAT + 2 * lane;
    atomicAdd(po,     wq * xv.x);
    atomicAdd(po + 1, wq * xv.y);
  }
}

extern "C" void kernel_launch(void* const* d_in, const int* in_sizes, int n_in,
                              void* d_out, int out_size, void* d_ws, size_t ws_size,
                              hipStream_t stream) {
  const float* x      = (const float*)d_in[0];
  const float* params = (const float*)d_in[1];
  const int*   sints  = (const int*)d_in[2];
  float* out = (float*)d_out;

  // zero output (harness poisons d_out; atomics need clean accumulators)
  const int n4 = OUT_NUM * DFEAT / 4;
  MH_zero_kernel<<<(n4 + 255) / 256, 256, 0, stream>>>((float4*)out, n4);

  // 8 waves/block, 8 k's per wave -> K / 64 blocks (exact)
  MH_hyper_kernel<<<KTOT / 64, 256, 0, stream>>>(x, params, sints, out);
}